// PagedAttention_57715770523804
// MI455X (gfx1250) — compile-verified
//
#include <hip/hip_runtime.h>
#include <math.h>

// CDNA5 / gfx1250 paged-attention decode.
// fp32 reference -> V_WMMA_F32_16X16X4_F32 for Q*K^T.
// Bandwidth-bound: 1 workgroup per (batch, kv-head), 8 wave32 waves flash-decode
// disjoint 16-position page tiles. K pages are DMA'd by the Tensor Data Mover
// (tensor_load_to_lds, TENSORcnt) into per-wave double buffers in LDS and
// consumed as WMMA A-fragments; V is streamed with coalesced b128 loads.

typedef __attribute__((ext_vector_type(2))) float        v2f;
typedef __attribute__((ext_vector_type(8))) float        v8f;
typedef __attribute__((ext_vector_type(4))) unsigned int v4u;
typedef __attribute__((ext_vector_type(8))) int          v8i;
typedef __attribute__((ext_vector_type(4))) int          v4i;

#define NEG_INF_F (-1e30f)

__device__ __forceinline__ float lane_sel4(float a0, float a1, float a2, float a3, int q) {
    return q == 0 ? a0 : (q == 1 ? a1 : (q == 2 ? a2 : (q == 3 ? a3 : 0.0f)));
}

// TDM: DMA one K page tile (16 rows x 128 f32, row stride 1024 f32) into LDS.
// D# per CDNA5 ISA ch.8: group0 = control/lds/global addr, group1 = dims/strides.
__device__ __forceinline__ void tdm_load_k_tile(unsigned int lds_byte, unsigned long long gaddr) {
    v4u g0 = {
        1u,                                              // count=1, user mode, no gather
        lds_byte,                                        // lds_addr (bytes)
        (unsigned int)gaddr,                             // global_addr[31:0]
        (unsigned int)(gaddr >> 32) | (2u << 30)         // global_addr[56:32] | type=2
    };
    v8i g1 = {
        (int)(2u << 16),        // wg_mask=0 | data_size=2 (4 bytes)
        (int)(128u << 16),      // atomic_barrier=0 | tensor_dim0[15:0]=128
        (int)(16u << 16),       // tensor_dim0[31:16]=0 | tensor_dim1[15:0]=16
        (int)(128u << 16),      // tensor_dim1[31:16]=0 | tile_dim0=128
        16,                     // tile_dim1=16 | tile_dim2=0
        1024,                   // tensor_dim0_stride[31:0]=1024 elements (Hkv*D)
        0,                      // tensor_dim0_stride[47:32]=0 | tensor_dim1_stride lo=0
        0                       // tensor_dim1_stride hi=0
    };
    v4i z4 = {0, 0, 0, 0};                      // groups 2/3 unused (2D tile)
    v8i z8 = {0, 0, 0, 0, 0, 0, 0, 0};          // extra group (6-arg toolchain form)
    __builtin_amdgcn_tensor_load_to_lds(g0, g1, z4, z4, z8, 0);
}

__global__ __launch_bounds__(256) void paged_attn_wmma_f32(
    const float* __restrict__ query,       // [B,H,D]
    const float* __restrict__ key,         // [B,Hkv,D]
    const float* __restrict__ value,       // [B,Hkv,D]
    const float* __restrict__ k_cache,     // [NB,16,Hkv,D]
    const float* __restrict__ v_cache,     // [NB,16,Hkv,D]
    const int*   __restrict__ block_table, // [B,256]
    const int*   __restrict__ context_lens,// [B]
    float*       __restrict__ out)         // [B,H,D]
{
    constexpr int H = 32, HKV = 8, G = 4, D = 128, BS = 16, MAXB = 256, NW = 8;
    constexpr int TILE = BS * D;                        // 2048 floats per K page tile
    constexpr float SCALE = 0.08838834764831845f;       // 1/sqrt(128)
    constexpr float LOG1E4_64 = 0.14391156457f;         // ln(10000)/64

    const int b    = blockIdx.x >> 3;
    const int kvh  = blockIdx.x & 7;
    const int tid  = threadIdx.x;
    const int wave = tid >> 5;
    const int lane = tid & 31;

    __shared__ float lds_kt[NW * 2 * TILE];  // per-wave double-buffered K tiles (128 KB)
    __shared__ float lds_q[16 * D];          // rows 0..3 roped Q heads, rows 4..15 zero
    __shared__ float lds_kn[D];              // roped new key
    __shared__ float lds_m[NW * G];
    __shared__ float lds_l[NW * G];
    float* lds_o = lds_kt;                   // merge scratch aliased onto dead K buffers

    const int   ctx  = context_lens[b];
    const int   last = ctx - 1;
    const float posf = (float)ctx;           // reference ropes with positions = context_lens

    // ---------------- RoPE q (4 heads) and k_new into LDS ----------------
    for (int i = tid; i < 12 * D; i += 256) lds_q[4 * D + i] = 0.0f;
    const float* qbase = query + ((size_t)b * H + (size_t)kvh * G) * D;
    for (int i = tid; i < G * D; i += 256) {
        int g = i >> 7, d = i & 127;
        int ii = (d < 64) ? d : d - 64;
        float f = posf * __expf(-(float)ii * LOG1E4_64);
        float c = __cosf(f), s = __sinf(f);
        float x     = qbase[g * D + d];
        float other = (d < 64) ? -qbase[g * D + d + 64] : qbase[g * D + d - 64];
        lds_q[i] = x * c + other * s;
    }
    if (tid < D) {
        int d = tid;
        int ii = (d < 64) ? d : d - 64;
        float f = posf * __expf(-(float)ii * LOG1E4_64);
        float c = __cosf(f), s = __sinf(f);
        const float* kp = key + ((size_t)b * HKV + kvh) * D;
        float x     = kp[d];
        float other = (d < 64) ? -kp[d + 64] : kp[d - 64];
        lds_kn[d] = x * c + other * s;
    }
    __syncthreads();

    // per-lane geometry for the WMMA fragments
    const int q4     = lane & 15;         // N column (query head) in B/C
    const int row    = lane & 15;         // M row (position in tile) in A
    const int cp     = (lane >> 4) << 1;  // K-pair column: lanes<16 -> {0,1}, >=16 -> {2,3}
    const int rowoff = (lane >> 4) << 3;  // C-layout: lane-half adds +8 to M
    const int dlo    = lane << 2;         // 4 d-values per lane for PV

    // ---- score of the (roped) new key vs each q head (tile-invariant) ----
    float pg0 = 0, pg1 = 0, pg2 = 0, pg3 = 0;
#pragma unroll
    for (int j = 0; j < 4; ++j) {
        float kn = lds_kn[dlo + j];
        pg0 += lds_q[0 * D + dlo + j] * kn;
        pg1 += lds_q[1 * D + dlo + j] * kn;
        pg2 += lds_q[2 * D + dlo + j] * kn;
        pg3 += lds_q[3 * D + dlo + j] * kn;
    }
#pragma unroll
    for (int off = 16; off >= 1; off >>= 1) {
        pg0 += __shfl_xor(pg0, off);
        pg1 += __shfl_xor(pg1, off);
        pg2 += __shfl_xor(pg2, off);
        pg3 += __shfl_xor(pg3, off);
    }
    const float sl_mine = lane_sel4(pg0, pg1, pg2, pg3, q4) * SCALE;

    // ---------------- flash-decode over page tiles (TDM double-buffered) ----------------
    float rm0 = NEG_INF_F, rm1 = NEG_INF_F, rm2 = NEG_INF_F, rm3 = NEG_INF_F;
    float rl0 = 0, rl1 = 0, rl2 = 0, rl3 = 0;
    float o0[4] = {0, 0, 0, 0}, o1[4] = {0, 0, 0, 0}, o2[4] = {0, 0, 0, 0}, o3[4] = {0, 0, 0, 0};

    const int*   bt        = block_table + (size_t)b * MAXB;
    const size_t rowstride = (size_t)HKV * D;       // 1024 floats
    const size_t blkstride = (size_t)BS * rowstride;
    const int    nTiles    = (ctx + BS - 1) / BS;
    const float* vnew      = value + ((size_t)b * HKV + kvh) * D + dlo;

    const unsigned int kbuf_base =
        (unsigned int)(size_t)(&lds_kt[(size_t)(wave * 2) * TILE]);   // LDS byte offset, buffer 0

    // prologue: DMA first tile for this wave
    if (wave < nTiles) {
        unsigned long long g = (unsigned long long)(size_t)(
            k_cache + (size_t)bt[wave] * blkstride + (size_t)kvh * D);
        tdm_load_k_tile(kbuf_base, g);
    }

    int pb = 0;
    for (int t = wave; t < nTiles; t += NW, pb ^= 1) {
        const int blk  = bt[t];
        const int base = t * BS;
        const float* vbase = v_cache + (size_t)blk * blkstride + (size_t)kvh * D + dlo;
        const bool hasNext = (t + NW) < nTiles;

        if (hasNext) {  // DMA next tile into the other buffer while we compute
            int nblk = bt[t + NW];
            unsigned long long g = (unsigned long long)(size_t)(
                k_cache + (size_t)nblk * blkstride + (size_t)kvh * D);
            asm volatile("" ::: "memory");
            tdm_load_k_tile(kbuf_base + (unsigned int)((pb ^ 1) * TILE * 4), g);
            __builtin_prefetch(v_cache + (size_t)nblk * blkstride + (size_t)kvh * D, 0, 0);
            __builtin_amdgcn_s_wait_tensorcnt((short)1);   // retire current tile's DMA
        } else {
            __builtin_amdgcn_s_wait_tensorcnt((short)0);
        }
        asm volatile("" ::: "memory");

        const float* kbuf = &lds_kt[(size_t)(wave * 2 + pb) * TILE + (size_t)row * D + cp];

        // scores^T tile (16 pos x 16 cols, 4 real) = K_tile(16x128) * Q^T(128x16)
        v8f C = {0, 0, 0, 0, 0, 0, 0, 0};
#pragma unroll
        for (int dc = 0; dc < 32; ++dc) {
            v2f a  = *(const v2f*)(kbuf + dc * 4);                 // A: 16x4 f32 (from LDS)
            v2f bq = *(const v2f*)(&lds_q[q4 * D + dc * 4 + cp]);  // B: 4x16 f32
            C = __builtin_amdgcn_wmma_f32_16x16x4_f32(false, a, false, bq, (short)0, C, false, false);
        }

        // mask + scale + tile max
        float p[8];
        float tmax = NEG_INF_F;
#pragma unroll
        for (int r = 0; r < 8; ++r) {
            int pos = base + r + rowoff;
            float s = C[r] * SCALE;
            if (pos == last) s = sl_mine;   // cache-update substitution (roped k_new)
            if (pos >= ctx)  s = NEG_INF_F;
            p[r] = s;
            tmax = fmaxf(tmax, s);
        }
        tmax = fmaxf(tmax, __shfl_xor(tmax, 16));
        float tm0 = __shfl(tmax, 0), tm1 = __shfl(tmax, 1), tm2 = __shfl(tmax, 2), tm3 = __shfl(tmax, 3);

        float nm0 = fmaxf(rm0, tm0), nm1 = fmaxf(rm1, tm1), nm2 = fmaxf(rm2, tm2), nm3 = fmaxf(rm3, tm3);
        float a0 = __expf(rm0 - nm0), a1 = __expf(rm1 - nm1), a2 = __expf(rm2 - nm2), a3 = __expf(rm3 - nm3);
        rm0 = nm0; rm1 = nm1; rm2 = nm2; rm3 = nm3;
#pragma unroll
        for (int j = 0; j < 4; ++j) { o0[j] *= a0; o1[j] *= a1; o2[j] *= a2; o3[j] *= a3; }
        rl0 *= a0; rl1 *= a1; rl2 *= a2; rl3 *= a3;

        const float nm_mine = lane_sel4(nm0, nm1, nm2, nm3, q4);
        float ls = 0.0f;
#pragma unroll
        for (int r = 0; r < 8; ++r) { p[r] = __expf(p[r] - nm_mine); ls += p[r]; }
        ls += __shfl_xor(ls, 16);
        rl0 += __shfl(ls, 0); rl1 += __shfl(ls, 1); rl2 += __shfl(ls, 2); rl3 += __shfl(ls, 3);

        // PV: coalesced float4 V rows, prob broadcast by shuffle
#pragma unroll
        for (int pp = 0; pp < 16; ++pp) {
            int pos = base + pp;
            if (pos < ctx) {  // wave-uniform
                const float4 v = (pos == last) ? *(const float4*)vnew
                                               : *(const float4*)(vbase + (size_t)pp * rowstride);
                const int sb = (pp < 8) ? 0 : 16;
                float w0 = __shfl(p[pp & 7], sb + 0);
                float w1 = __shfl(p[pp & 7], sb + 1);
                float w2 = __shfl(p[pp & 7], sb + 2);
                float w3 = __shfl(p[pp & 7], sb + 3);
                o0[0] += w0 * v.x; o0[1] += w0 * v.y; o0[2] += w0 * v.z; o0[3] += w0 * v.w;
                o1[0] += w1 * v.x; o1[1] += w1 * v.y; o1[2] += w1 * v.z; o1[3] += w1 * v.w;
                o2[0] += w2 * v.x; o2[1] += w2 * v.y; o2[2] += w2 * v.z; o2[3] += w2 * v.w;
                o3[0] += w3 * v.x; o3[1] += w3 * v.y; o3[2] += w3 * v.z; o3[3] += w3 * v.w;
            }
        }
    }

    // ---------------- cross-wave LSE merge (reuses K-buffer LDS) ----------------
    __syncthreads();   // all waves done streaming K tiles; lds_kt is dead
    float* po = &lds_o[(size_t)(wave * G) * D];
#pragma unroll
    for (int j = 0; j < 4; ++j) {
        po[0 * D + dlo + j] = o0[j];
        po[1 * D + dlo + j] = o1[j];
        po[2 * D + dlo + j] = o2[j];
        po[3 * D + dlo + j] = o3[j];
    }
    if (lane == 0) {
        lds_m[wave * G + 0] = rm0; lds_m[wave * G + 1] = rm1;
        lds_m[wave * G + 2] = rm2; lds_m[wave * G + 3] = rm3;
        lds_l[wave * G + 0] = rl0; lds_l[wave * G + 1] = rl1;
        lds_l[wave * G + 2] = rl2; lds_l[wave * G + 3] = rl3;
    }
    __syncthreads();

#pragma unroll
    for (int e = 0; e < 2; ++e) {
        int idx = tid + e * 256;          // 512 outputs: 4 heads x 128 d
        int q = idx >> 7, d = idx & 127;
        float M = NEG_INF_F;
#pragma unroll
        for (int w = 0; w < NW; ++w) M = fmaxf(M, lds_m[w * G + q]);
        float L = 0.0f, acc = 0.0f;
#pragma unroll
        for (int w = 0; w < NW; ++w) {
            float al = __expf(lds_m[w * G + q] - M);
            L   += lds_l[w * G + q] * al;
            acc += lds_o[(size_t)(w * G + q) * D + d] * al;
        }
        out[((size_t)b * H + (size_t)kvh * G + q) * D + d] = acc / L;
    }
}

extern "C" void kernel_launch(void* const* d_in, const int* in_sizes, int n_in,
                              void* d_out, int out_size, void* d_ws, size_t ws_size,
                              hipStream_t stream) {
    (void)in_sizes; (void)n_in; (void)out_size; (void)d_ws; (void)ws_size;
    const float* query        = (const float*)d_in[0];
    const float* key          = (const float*)d_in[1];
    const float* value        = (const float*)d_in[2];
    const float* k_cache      = (const float*)d_in[3];
    const float* v_cache      = (const float*)d_in[4];
    const int*   block_table  = (const int*)d_in[5];
    const int*   context_lens = (const int*)d_in[6];
    // d_in[7] = block_size scalar (16, compile-time constant in kernel)

    paged_attn_wmma_f32<<<dim3(32 * 8), dim3(256), 0, stream>>>(
        query, key, value, k_cache, v_cache, block_table, context_lens, (float*)d_out);
}